// GCN_36361193128467
// MI455X (gfx1250) — compile-verified
//
#include <hip/hip_runtime.h>
#include <hip/hip_bf16.h>
#include <math.h>

typedef __attribute__((ext_vector_type(2))) float v2f;
typedef __attribute__((ext_vector_type(8))) float v8f;

#define DIN   1022
#define HFEAT 10
#define DOUTC 2
#define PADC  16   // padded feature stride (multiple of 16 for WMMA N-dim)

// ---------------- small utility kernels ----------------

__global__ void fill_kernel(float* __restrict__ p, float v, int n) {
  int i = blockIdx.x * blockDim.x + threadIdx.x;
  if (i < n) p[i] = v;
}

__global__ void deg_kernel(const long long* __restrict__ dst, float* __restrict__ deg, int E) {
  int e = blockIdx.x * blockDim.x + threadIdx.x;
  if (e < E) atomicAdd(&deg[(int)dst[e]], 1.0f);
}

__global__ void rsqrt_kernel(float* __restrict__ p, int n) {
  int i = blockIdx.x * blockDim.x + threadIdx.x;
  if (i < n) p[i] = rsqrtf(p[i]);
}

// pad weight [K,H] (row-major) -> [K,16] with zero columns >= H
__global__ void padw_kernel(const float* __restrict__ w, float* __restrict__ wp, int K, int H) {
  int i = blockIdx.x * blockDim.x + threadIdx.x;
  if (i >= K * PADC) return;
  int k = i >> 4, c = i & 15;
  wp[i] = (c < H) ? w[k * H + c] : 0.0f;
}

// ---------------- WMMA fp32 GEMM: C[M x 16] = A[M x K] @ Bp[K x 16] ----------------
// One wave32 computes one 16x16 tile. A row stride = lda, C row stride = PADC.
__global__ void gemm_wmma_kernel(const float* __restrict__ A, const float* __restrict__ Bp,
                                 float* __restrict__ C, int M, int numTiles, int K, int lda) {
  int wave = (int)((blockIdx.x * blockDim.x + threadIdx.x) >> 5);
  int lane = threadIdx.x & 31;
  if (wave >= numTiles) return;

  int row0  = wave * 16;
  int hi    = lane >> 4;          // 0: lanes 0-15, 1: lanes 16-31
  int khalf = hi * 2;             // K sub-offset within a 4-chunk
  int ncol  = lane & 15;

  int mrow = row0 + (lane & 15);
  if (mrow >= M) mrow = M - 1;    // clamp (defensive; N % 16 == 0 here)
  const float* arow = A + (size_t)mrow * lda;

  v8f acc = {};
  int Kmain = K & ~3;
  for (int kb = 0; kb < Kmain; kb += 4) {
    v2f a, b;
    a.x = arow[kb + khalf];
    a.y = arow[kb + khalf + 1];
    b.x = Bp[(size_t)(kb + khalf) * PADC + ncol];
    b.y = Bp[(size_t)(kb + khalf + 1) * PADC + ncol];
    acc = __builtin_amdgcn_wmma_f32_16x16x4_f32(false, a, false, b, (short)0, acc, false, false);
  }
  if (Kmain < K) {                // K % 4 in {1,2,3}
    v2f a = {}, b = {};
    int k0 = Kmain + khalf, k1 = k0 + 1;
    if (k0 < K) { a.x = arow[k0]; b.x = Bp[(size_t)k0 * PADC + ncol]; }
    if (k1 < K) { a.y = arow[k1]; b.y = Bp[(size_t)k1 * PADC + ncol]; }
    acc = __builtin_amdgcn_wmma_f32_16x16x4_f32(false, a, false, b, (short)0, acc, false, false);
  }

  // C/D layout: VGPR r -> row (row0 + 8*hi + r), col = ncol
  int rbase = row0 + hi * 8;
  float* cptr = C + (size_t)rbase * PADC + ncol;
#pragma unroll
  for (int r = 0; r < 8; ++r) {
    if (rbase + r < M) cptr[(size_t)r * PADC] = acc[r];
  }
}

// ---------------- edge scatter-add: agg[dst] += h[src] * dinv[src]*dinv[dst] ----------------
template <int F>
__global__ void edgeagg_kernel(const long long* __restrict__ src, const long long* __restrict__ dst,
                               const float* __restrict__ dinv, const float* __restrict__ h,
                               float* __restrict__ agg, int E) {
  int e = blockIdx.x * blockDim.x + threadIdx.x;
  if (e >= E) return;
  int s = (int)src[e], d = (int)dst[e];
  float c = dinv[s] * dinv[d];
  const float* hp = h + (size_t)s * PADC;
  float* ap = agg + (size_t)d * PADC;
#pragma unroll
  for (int f = 0; f < F; ++f) atomicAdd(&ap[f], hp[f] * c);
}

// ---------------- self-loop + bias (+relu); writes zero into pad columns ----------------
__global__ void post_kernel(const float* __restrict__ h, float* __restrict__ agg,
                            const float* __restrict__ dinv, const float* __restrict__ bias,
                            int N, int F, int relu) {
  int i = blockIdx.x * blockDim.x + threadIdx.x;
  int node = i >> 4, f = i & 15;
  if (node >= N) return;
  float v = 0.0f;
  if (f < F) {
    float di = dinv[node];
    v = agg[(size_t)node * PADC + f] + h[(size_t)node * PADC + f] * di * di + bias[f];
    if (relu) v = fmaxf(v, 0.0f);
  }
  agg[(size_t)node * PADC + f] = v;
}

// ---------------- final layer combine + log_softmax over 2 classes ----------------
__global__ void final_kernel(const float* __restrict__ h, const float* __restrict__ agg,
                             const float* __restrict__ dinv, const float* __restrict__ b3,
                             float* __restrict__ out, int N) {
  int i = blockIdx.x * blockDim.x + threadIdx.x;
  if (i >= N) return;
  float di = dinv[i];
  float d2 = di * di;
  float v0 = agg[(size_t)i * PADC + 0] + h[(size_t)i * PADC + 0] * d2 + b3[0];
  float v1 = agg[(size_t)i * PADC + 1] + h[(size_t)i * PADC + 1] * d2 + b3[1];
  float m  = fmaxf(v0, v1);
  float lse = m + logf(expf(v0 - m) + expf(v1 - m));
  out[(size_t)i * 2 + 0] = v0 - lse;
  out[(size_t)i * 2 + 1] = v1 - lse;
}

// ---------------- host-side launch ----------------

extern "C" void kernel_launch(void* const* d_in, const int* in_sizes, int n_in,
                              void* d_out, int out_size, void* d_ws, size_t ws_size,
                              hipStream_t stream) {
  const float*     x  = (const float*)d_in[0];
  const long long* ei = (const long long*)d_in[1];   // int64 edge_index [2, E]
  const float*     W1 = (const float*)d_in[2];
  const float*     b1 = (const float*)d_in[3];
  const float*     W2 = (const float*)d_in[4];
  const float*     b2 = (const float*)d_in[5];
  const float*     W3 = (const float*)d_in[6];
  const float*     b3 = (const float*)d_in[7];
  float* out = (float*)d_out;

  const int N = in_sizes[0] / DIN;
  const int E = in_sizes[1] / 2;
  const long long* src = ei;
  const long long* dst = ei + E;

  // workspace carve-out
  char*  ws  = (char*)d_ws;
  size_t off = 0;
  auto carve = [&](size_t bytes) -> void* {
    void* p = ws + off;
    off += (bytes + 255) & ~(size_t)255;
    return p;
  };
  float* b0   = (float*)carve((size_t)N * PADC * sizeof(float));  // GEMM outputs h
  float* bagg = (float*)carve((size_t)N * PADC * sizeof(float));  // aggregation / layer outputs
  float* dinv = (float*)carve((size_t)N * sizeof(float));
  float* W1p  = (float*)carve((size_t)DIN * PADC * sizeof(float));
  float* W2p  = (float*)carve((size_t)HFEAT * PADC * sizeof(float));
  float* W3p  = (float*)carve((size_t)HFEAT * PADC * sizeof(float));

  const int TB = 256;
  const int nPad  = N * PADC;
  const int gN    = (N + TB - 1) / TB;
  const int gE    = (E + TB - 1) / TB;
  const int gPad  = (nPad + TB - 1) / TB;

  // degrees with self-loop: deg = 1 + sum over incoming edges; dinv = rsqrt(deg)
  fill_kernel<<<gN, TB, 0, stream>>>(dinv, 1.0f, N);
  deg_kernel<<<gE, TB, 0, stream>>>(dst, dinv, E);
  rsqrt_kernel<<<gN, TB, 0, stream>>>(dinv, N);

  // pad weights to 16 columns
  padw_kernel<<<(DIN * PADC + TB - 1) / TB, TB, 0, stream>>>(W1, W1p, DIN, HFEAT);
  padw_kernel<<<(HFEAT * PADC + TB - 1) / TB, TB, 0, stream>>>(W2, W2p, HFEAT, HFEAT);
  padw_kernel<<<(HFEAT * PADC + TB - 1) / TB, TB, 0, stream>>>(W3, W3p, HFEAT, DOUTC);

  const int numTiles = (N + 15) / 16;
  const int wavesPerBlock = TB / 32;
  const int gGemm = (numTiles + wavesPerBlock - 1) / wavesPerBlock;

  // ---- layer 1 ----
  gemm_wmma_kernel<<<gGemm, TB, 0, stream>>>(x, W1p, b0, N, numTiles, DIN, DIN);
  fill_kernel<<<gPad, TB, 0, stream>>>(bagg, 0.0f, nPad);
  edgeagg_kernel<HFEAT><<<gE, TB, 0, stream>>>(src, dst, dinv, b0, bagg, E);
  post_kernel<<<gPad, TB, 0, stream>>>(b0, bagg, dinv, b1, N, HFEAT, 1);

  // ---- layer 2 ----
  gemm_wmma_kernel<<<gGemm, TB, 0, stream>>>(bagg, W2p, b0, N, numTiles, HFEAT, PADC);
  fill_kernel<<<gPad, TB, 0, stream>>>(bagg, 0.0f, nPad);
  edgeagg_kernel<HFEAT><<<gE, TB, 0, stream>>>(src, dst, dinv, b0, bagg, E);
  post_kernel<<<gPad, TB, 0, stream>>>(b0, bagg, dinv, b2, N, HFEAT, 1);

  // ---- layer 3 + log_softmax ----
  gemm_wmma_kernel<<<gGemm, TB, 0, stream>>>(bagg, W3p, b0, N, numTiles, HFEAT, PADC);
  fill_kernel<<<gPad, TB, 0, stream>>>(bagg, 0.0f, nPad);
  edgeagg_kernel<DOUTC><<<gE, TB, 0, stream>>>(src, dst, dinv, b0, bagg, E);
  final_kernel<<<gN, TB, 0, stream>>>(b0, bagg, dinv, b3, out, N);
}